// Proposal_83038897701555
// MI455X (gfx1250) — compile-verified
//
#include <hip/hip_runtime.h>
#include <cstdint>

#define H_     76
#define W_     128
#define A_     9
#define HW_    (H_*W_)        // 9728
#define N_     (H_*W_*A_)     // 87552
#define PRE_   6000
#define POST_  1000
#define NMSTH  0.7f
#define COLW   188            // ceil(6000/32)
#define NPAD   6016           // 188*32

// ---- workspace layout, in 32-bit words (all 16B-aligned regions) ----
#define OFF_BOXES   0u          // N_*4 floats            = 350208 words
#define OFF_FGKEY   350208u     // N_ uint64 keys         = 175104 words
#define OFF_SBOX    525312u     // NPAD*4 floats          =  24064 words
#define OFF_MASK    549376u     // NPAD*COLW uint words   = 1131008 words
#define OFF_KEEP    1680384u    // COLW uint words

// Anchors from generate_anchors(16) with scales {8,16,32}, ratios {0.5,1,2}
__constant__ float kAnchors[9][4] = {
  { -84.f,  -40.f,  99.f,  55.f},
  {-176.f,  -88.f, 191.f, 103.f},
  {-360.f, -184.f, 375.f, 199.f},
  { -56.f,  -56.f,  71.f,  71.f},
  {-120.f, -120.f, 135.f, 135.f},
  {-248.f, -248.f, 263.f, 263.f},
  { -36.f,  -80.f,  51.f,  95.f},
  { -80.f, -168.f,  95.f, 183.f},
  {-168.f, -344.f, 183.f, 359.f},
};

// ---------- CDNA5 async-to-LDS helpers ----------
typedef int v4i_ __attribute__((vector_size(16)));
typedef __attribute__((address_space(1))) v4i_ gv4i_t;
typedef __attribute__((address_space(3))) v4i_ lv4i_t;

__device__ __forceinline__ void async_copy16(void* lds_dst, const void* gsrc) {
#if __has_builtin(__builtin_amdgcn_global_load_async_to_lds_b128)
  v4i_* g = (v4i_*)gsrc;        // C-cast drops const, keeps generic AS
  v4i_* l = (v4i_*)lds_dst;
  __builtin_amdgcn_global_load_async_to_lds_b128(
      (gv4i_t*)g, (lv4i_t*)l, 0, 0);
#else
  *(float4*)lds_dst = *(const float4*)gsrc;   // synchronous fallback
#endif
}

__device__ __forceinline__ void async_wait0() {
#if __has_builtin(__builtin_amdgcn_s_wait_asynccnt)
  __builtin_amdgcn_s_wait_asynccnt(0);
#else
  asm volatile("s_wait_asynccnt 0" ::: "memory");
#endif
}

// ---------------- Kernel 1: anchor/box decode + u64 sort keys ----------------
__global__ __launch_bounds__(256)
void k_decode(const float* __restrict__ scores,
              const float* __restrict__ deltas,
              const float* __restrict__ iminfo,
              float* __restrict__ boxes,
              unsigned long long* __restrict__ fgkey) {
  int idx = blockIdx.x * 256 + threadIdx.x;
  if (idx >= N_) return;
  int a   = idx % A_;
  int pix = idx / A_;
  int wx  = pix % W_;
  int hy  = pix / W_;
  int p   = pix;                       // h*W + w
  float sx = (float)(wx * 16), sy = (float)(hy * 16);
  const float* an = kAnchors[a];
  float x1 = an[0] + sx, y1 = an[1] + sy, x2 = an[2] + sx, y2 = an[3] + sy;
  float aw = x2 - x1 + 1.0f, ah = y2 - y1 + 1.0f;
  float acx = x1 + 0.5f * aw, acy = y1 + 0.5f * ah;
  float d0 = deltas[(4*a+0)*HW_ + p];
  float d1 = deltas[(4*a+1)*HW_ + p];
  float d2 = deltas[(4*a+2)*HW_ + p];
  float d3 = deltas[(4*a+3)*HW_ + p];
  float pcx = d0 * aw + acx, pcy = d1 * ah + acy;
  float pw  = expf(d2) * aw, ph  = expf(d3) * ah;
  float xmax = iminfo[1] - 1.0f, ymax = iminfo[0] - 1.0f;
  float bx1 = fminf(fmaxf(pcx - 0.5f*pw, 0.0f), xmax);
  float by1 = fminf(fmaxf(pcy - 0.5f*ph, 0.0f), ymax);
  float bx2 = fminf(fmaxf(pcx + 0.5f*pw, 0.0f), xmax);
  float by2 = fminf(fmaxf(pcy + 0.5f*ph, 0.0f), ymax);
  float4 b = make_float4(bx1, by1, bx2, by2);
  ((float4*)boxes)[idx] = b;
  // score >= 0 => uint bit pattern is order-preserving; low bits break ties
  // in favor of the smaller index (matches lax.top_k stability).
  unsigned u = __float_as_uint(scores[(A_ + a)*HW_ + p]);
  fgkey[idx] = (((unsigned long long)u) << 17) | (unsigned)(N_ - 1 - idx);
}

// ---------------- Kernel 2: exact rank-by-counting top-6000 ----------------
__global__ __launch_bounds__(256)
void k_rank(const unsigned long long* __restrict__ fgkey,
            const float* __restrict__ boxes,
            float* __restrict__ sboxes) {
  __shared__ alignas(16) unsigned long long tile[512];
  int i = blockIdx.x * 256 + threadIdx.x;
  unsigned long long ki = fgkey[i];
  int rank = 0;
  for (int t = 0; t < N_ / 512; ++t) {
    __syncthreads();
    // 256 lanes x 16B async copies fill the 4KB tile (CDNA5 async path)
    async_copy16(&tile[2 * threadIdx.x], &fgkey[t * 512 + 2 * threadIdx.x]);
    async_wait0();
    __syncthreads();
    const ulonglong2* t2 = (const ulonglong2*)tile;
#pragma unroll 8
    for (int k = 0; k < 256; ++k) {
      ulonglong2 v = t2[k];            // ds_load_b128, broadcast (no conflicts)
      rank += (v.x > ki);
      rank += (v.y > ki);
    }
  }
  if (rank < PRE_) {
    ((float4*)sboxes)[rank] = ((const float4*)boxes)[i];
  }
}

// ---------------- Kernel 3: 32x32 suppression bitmask (upper triangle) -------
__global__ __launch_bounds__(32)
void k_mask(const float* __restrict__ sboxes, unsigned* __restrict__ mask) {
  int rb = blockIdx.y, cb = blockIdx.x;
  if (cb < rb) return;
  __shared__ alignas(16) float cbox[32 * 4];
  int t = threadIdx.x;
  async_copy16(&cbox[t * 4], &sboxes[(cb * 32 + t) * 4]);
  async_wait0();
  __syncthreads();
  int ig = rb * 32 + t;
  float4 r = ((const float4*)sboxes)[ig];
  float ax1 = r.x, ay1 = r.y, ax2 = r.z, ay2 = r.w;
  float aarea = (ax2 - ax1 + 1.0f) * (ay2 - ay1 + 1.0f);
  unsigned bits = 0u;
#pragma unroll
  for (int c = 0; c < 32; ++c) {
    int jg = cb * 32 + c;
    float bx1 = cbox[c*4+0], by1 = cbox[c*4+1], bx2 = cbox[c*4+2], by2 = cbox[c*4+3];
    float iw = fminf(ax2, bx2) - fmaxf(ax1, bx1) + 1.0f;
    float ih = fminf(ay2, by2) - fmaxf(ay1, by1) + 1.0f;
    iw = fmaxf(iw, 0.0f); ih = fmaxf(ih, 0.0f);
    float inter = iw * ih;
    float barea = (bx2 - bx1 + 1.0f) * (by2 - by1 + 1.0f);
    float iou = inter / (aarea + barea - inter);
    if (jg > ig && iou > NMSTH) bits |= (1u << c);
  }
  if (ig < PRE_) mask[(size_t)ig * COLW + cb] = bits;
}

// -------- Kernel 4: sequential greedy scan, one wave32, keep-words in VGPRs --
__global__ __launch_bounds__(32)
void k_scan(const unsigned* __restrict__ mask, unsigned* __restrict__ keepw) {
  int lane = threadIdx.x;
  unsigned kw[6];
#pragma unroll
  for (int s = 0; s < 6; ++s) {
    int wl = lane + 32 * s;
    unsigned v = (wl < COLW) ? 0xFFFFFFFFu : 0u;
    if (wl == COLW - 1) v = 0x0000FFFFu;   // bits for i>=6000 are invalid
    kw[s] = v;
  }
  for (int i = 0; i < PRE_; ++i) {
    __builtin_prefetch(mask + (size_t)(i + 1) * COLW + lane, 0, 0);
    int w = i >> 5, slot = w >> 5, src = w & 31;
    unsigned vv;
    switch (slot) {                       // slot/src uniform across the wave
      case 0: vv = __shfl(kw[0], src, 32); break;
      case 1: vv = __shfl(kw[1], src, 32); break;
      case 2: vv = __shfl(kw[2], src, 32); break;
      case 3: vv = __shfl(kw[3], src, 32); break;
      case 4: vv = __shfl(kw[4], src, 32); break;
      default: vv = __shfl(kw[5], src, 32); break;
    }
    if ((vv >> (i & 31)) & 1u) {          // uniform branch
      const unsigned* row = mask + (size_t)i * COLW;
#pragma unroll
      for (int s = 0; s < 6; ++s) {
        int wl = lane + 32 * s;
        if (wl >= w && wl < COLW) kw[s] &= ~row[wl];  // only j>i words written
      }
    }
  }
#pragma unroll
  for (int s = 0; s < 6; ++s) {
    int wl = lane + 32 * s;
    if (wl < COLW) keepw[wl] = kw[s];
  }
}

// -------- Kernel 5: rank kept (then suppressed) entries, emit first 1000 -----
__global__ __launch_bounds__(256)
void k_select(const unsigned* __restrict__ keepw,
              const float* __restrict__ sboxes,
              float* __restrict__ out) {
  __shared__ unsigned kws[COLW];
  __shared__ unsigned pre[COLW + 1];
  int tid = threadIdx.x;
  if (tid < COLW) kws[tid] = keepw[tid];
  __syncthreads();
  if (tid == 0) {
    unsigned acc = 0;
    for (int w = 0; w < COLW; ++w) { pre[w] = acc; acc += __popc(kws[w]); }
    pre[COLW] = acc;
  }
  __syncthreads();
  int total = (int)pre[COLW];
  for (int i = tid; i < PRE_; i += 256) {
    int w = i >> 5, b = i & 31;
    unsigned word = kws[w];
    int below = (int)pre[w] + __popc(word & ((1u << b) - 1u));
    int rank = ((word >> b) & 1u) ? below : total + (i - below);
    if (rank < POST_) ((float4*)out)[rank] = ((const float4*)sboxes)[i];
  }
}

extern "C" void kernel_launch(void* const* d_in, const int* in_sizes, int n_in,
                              void* d_out, int out_size, void* d_ws, size_t ws_size,
                              hipStream_t stream) {
  const float* scores = (const float*)d_in[0];
  const float* deltas = (const float*)d_in[1];
  const float* iminfo = (const float*)d_in[2];
  float* ws = (float*)d_ws;
  float* boxes  = ws + OFF_BOXES;
  unsigned long long* fgkey = (unsigned long long*)(ws + OFF_FGKEY);
  float* sboxes = ws + OFF_SBOX;
  unsigned* mask  = (unsigned*)(ws + OFF_MASK);
  unsigned* keepw = (unsigned*)(ws + OFF_KEEP);
  float* out = (float*)d_out;

  k_decode<<<N_ / 256, 256, 0, stream>>>(scores, deltas, iminfo, boxes, fgkey);
  k_rank  <<<N_ / 256, 256, 0, stream>>>(fgkey, boxes, sboxes);
  k_mask  <<<dim3(COLW, COLW), 32, 0, stream>>>(sboxes, mask);
  k_scan  <<<1, 32, 0, stream>>>(mask, keepw);
  k_select<<<1, 256, 0, stream>>>(keepw, sboxes, out);
}